// EntityContext_63754494542685
// MI455X (gfx1250) — compile-verified
//
#include <hip/hip_runtime.h>
#include <hip/hip_bf16.h>

typedef __attribute__((ext_vector_type(2))) float v2f;
typedef __attribute__((ext_vector_type(4))) float v4f;
typedef __attribute__((ext_vector_type(8))) float v8f;

constexpr int BB     = 8192;
constexpr int MAXE   = 64;
constexpr int NE     = 65;      // MAXE+1
constexpr int DD     = 256;
constexpr int HIDN   = 256;
constexpr int MAXLEN = 25;
constexpr int PS     = 260;     // padded LDS stride (bank-conflict free, 16B aligned)
constexpr float EPSV = 1e-20f;

// output section offsets (flat float elements, tuple return order)
constexpr long O_ET   = 0;
constexpr long O_EIDX = O_ET   + (long)BB * 2;
constexpr long O_ELEN = O_EIDX + (long)BB * NE;
constexpr long O_X    = O_ELEN + (long)BB * MAXLEN;
constexpr long O_E2   = O_X    + (long)BB * DD;
constexpr long O_N    = O_E2   + (long)BB * NE * DD;
constexpr long O_D    = O_N    + (long)BB;
constexpr long O_NULL = O_D    + (long)BB * NE;
constexpr long O_LOOK = O_NULL + (long)BB * DD;

__device__ inline float wave_sum(float v) {
#pragma unroll
  for (int o = 16; o; o >>= 1) v += __shfl_xor(v, o, 32);
  return v;
}

__device__ inline v4f nt_load4(const float* p) {
  return __builtin_nontemporal_load((const v4f*)p);
}
__device__ inline void nt_store4(float* p, v4f v) {
  __builtin_nontemporal_store(v, (v4f*)p);
}

// 16x16 tile of  A(16xK, LDS row-major, stride `as`) @ W(NxK row-major)^T
// via V_WMMA_F32_16X16X4_F32 accumulation over K. W stays on the cached path
// (hot in L2: each W is only 256KB, reused by all 512 workgroups).
__device__ inline v8f gemm_tile(const float* A, int as,
                                const float* __restrict__ W, int wk,
                                int n0, int K) {
  const int lane = threadIdx.x & 31;
  const int half = lane >> 4;      // K-phase select (A: K0/K1 vs K2/K3)
  const int lm   = lane & 15;      // M for A, N for B
  v8f acc = {};
  const float* ap = A + lm * as + 2 * half;
  const float* bp = W + (size_t)(n0 + lm) * wk + 2 * half;
#pragma unroll 8
  for (int k = 0; k < K; k += 4) {
    v2f a = *(const v2f*)(ap + k);
    v2f b = *(const v2f*)(bp + k);
    acc = __builtin_amdgcn_wmma_f32_16x16x4_f32(false, a, false, b,
                                                (short)0, acc, false, false);
  }
  return acc;
}

__global__ __launch_bounds__(256, 1)
void entity_cell_fused(
    const float* __restrict__ h, const float* __restrict__ E,
    const int* __restrict__ n_entities, const float* __restrict__ e_dists,
    const float* __restrict__ null_context, const unsigned char* __restrict__ e_lookup,
    const int* __restrict__ e_t, const int* __restrict__ e_idx,
    const unsigned char* __restrict__ final_tok, const float* __restrict__ init_noise,
    const float* __restrict__ W_R, const float* __restrict__ W_ctx,
    const float* __restrict__ lam, const float* __restrict__ W_L,
    const float* __restrict__ b_L, const float* __restrict__ mean_init,
    const float* __restrict__ W_forget, const float* __restrict__ W_input,
    const float* __restrict__ W_X, const float* __restrict__ W_Xnull,
    float* __restrict__ out)
{
  __shared__ float sm_h[16 * PS];     // h tile          (WMMA A)
  __shared__ float sm_pf[16 * PS];    // proj_f -> null_new overlay (WMMA A in phase 4)
  __shared__ float sm_ce[16 * PS];    // curr_e / E2 row (WMMA A)
  __shared__ float sm_i[16 * 256];    // input gate rows
  __shared__ float sm_proj[16 * 256]; // ctx projection rows
  __shared__ int   se1[16];
  __shared__ int   sem[16];

  const int tid  = threadIdx.x;
  const int lane = tid & 31;
  const int wave = tid >> 5;
  const int b0   = blockIdx.x * 16;
  const float lam0 = lam[0];

  // ---- Phase 1: stage h tile into LDS (read-once -> NT loads) ----
  for (int idx = tid * 4; idx < 16 * 256; idx += 256 * 4) {
    int r = idx >> 8, c = idx & 255;
    v4f v = nt_load4(h + (size_t)(b0 + r) * HIDN + c);
    *(v4f*)(sm_h + r * PS + c) = v;
  }
  __syncthreads();

  // ---- Phase 2: proj_f / i / proj via WMMA (48 tiles over 8 waves) ----
  for (int t = wave; t < 48; t += 8) {
    int g  = t >> 4;
    int n0 = (t & 15) << 4;
    const float* W = (g == 0) ? W_forget : (g == 1) ? W_input : W_ctx;
    float* dst     = (g == 0) ? sm_pf    : (g == 1) ? sm_i    : sm_proj;
    int dsd        = (g == 0) ? PS : 256;
    v8f c = gemm_tile(sm_h, PS, W, HIDN, n0, HIDN);
    int half = lane >> 4, lm = lane & 15;
#pragma unroll
    for (int r = 0; r < 8; ++r)
      dst[(r + 8 * half) * dsd + n0 + lm] = c[r];
  }
  // out_e_t = h @ W_R^T  (16x2, tiny -> VALU)
  if (tid < 32) {
    int b = tid >> 1, r = tid & 1;
    float s = 0.f;
    for (int k = 0; k < HIDN; ++k) s += sm_h[b * PS + k] * W_R[r * HIDN + k];
    out[O_ET + (size_t)(b0 + b) * 2 + r] = s;
  }
  __syncthreads();

  // ---- Phase 3: per-row entity update (2 rows per wave, 32 lanes over D) ----
  for (int rr = 0; rr < 2; ++rr) {
    int b  = wave * 2 + rr;
    int bg = b0 + b;
    int e1 = e_idx[bg] + 1;
    bool em   = (e_t[bg] == 1);
    bool fin  = final_tok[bg] != 0;
    bool had  = e_lookup[(size_t)bg * (MAXE + 2) + e1] != 0;
    bool newe = em && !had;
    int j0 = lane * 8;

    v4f nz0 = nt_load4(init_noise + (size_t)bg * DD + j0);
    v4f nz1 = nt_load4(init_noise + (size_t)bg * DD + j0 + 4);
    float z[8], cur[8];
    float ss = 0.f;
#pragma unroll
    for (int j = 0; j < 8; ++j) {
      float nz = (j < 4) ? nz0[j & 3] : nz1[j & 3];
      float v = mean_init[j0 + j] + nz * 1e-4f;
      z[j] = v; ss += v * v;
    }
    ss = wave_sum(ss);
    float zs = newe ? (1.f / sqrtf(ss)) : 0.f;

    v4f er0 = nt_load4(E + ((size_t)bg * NE + e1) * DD + j0);
    v4f er1 = nt_load4(E + ((size_t)bg * NE + e1) * DD + j0 + 4);
    float dotf = 0.f;
#pragma unroll
    for (int j = 0; j < 8; ++j) {
      float ev = (j < 4) ? er0[j & 3] : er1[j & 3];
      float c = ev + z[j] * zs;
      cur[j] = c;
      dotf += c * sm_pf[b * PS + j0 + j];
    }
    dotf = wave_sum(dotf);
    float f = 1.f / (1.f + __expf(-dotf));

    float rowv[8];
    float rn = 0.f;
#pragma unroll
    for (int j = 0; j < 8; ++j) {
      float v = cur[j] * (1.f - f) + f * sm_i[b * 256 + j0 + j];
      rowv[j] = v; rn += v * v;
    }
    rn = wave_sum(rn);
    float inv = 1.f / (sqrtf(rn) + EPSV);

    v4f nc0 = nt_load4(null_context + (size_t)bg * DD + j0);
    v4f nc1 = nt_load4(null_context + (size_t)bg * DD + j0 + 4);
    v4f nnv0, nnv1;
#pragma unroll
    for (int j = 0; j < 8; ++j) {
      float e2v = em ? rowv[j] * inv : cur[j];
      sm_ce[b * PS + j0 + j] = e2v;
      float ncv = (j < 4) ? nc0[j & 3] : nc1[j & 3];
      float nv = fin ? sm_h[b * PS + j0 + j] : (em ? e2v : ncv);
      if (j < 4) nnv0[j & 3] = nv; else nnv1[j & 3] = nv;
    }
    nt_store4(out + O_NULL + (size_t)bg * DD + j0,     nnv0);
    nt_store4(out + O_NULL + (size_t)bg * DD + j0 + 4, nnv1);
    // overlay null_new onto proj_f rows (row-private to this wave; proj_f consumed)
#pragma unroll
    for (int j = 0; j < 8; ++j)
      sm_pf[b * PS + j0 + j] = (j < 4) ? nnv0[j & 3] : nnv1[j & 3];

    // lookup_new (66 bools as 0/1) + n_new + stash scalars
    for (int e = lane; e < MAXE + 2; e += 32) {
      bool v = (e == e1) || (e_lookup[(size_t)bg * (MAXE + 2) + e] != 0);
      out[O_LOOK + (size_t)bg * (MAXE + 2) + e] = v ? 1.f : 0.f;
    }
    if (lane == 0) {
      ((int*)out)[O_N + bg] = n_entities[bg] + (newe ? 1 : 0);
      se1[b] = e1;
      sem[b] = em ? 1 : 0;
    }
  }
  __syncthreads();

  // ---- Phase 4: out_x via dual WMMA GEMMs + per-row select; out_e_len ----
  {
    const int half = lane >> 4, lm = lane & 15;
    for (int t = wave; t < 16; t += 8) {
      int n0 = t << 4;
      v8f c1 = gemm_tile(sm_ce, PS, W_X,     DD, n0, DD);  // curr_e  @ W_X^T
      v8f c2 = gemm_tile(sm_pf, PS, W_Xnull, DD, n0, DD);  // null_new@ W_Xnull^T
#pragma unroll
      for (int r = 0; r < 8; ++r) {
        int m = r + 8 * half;
        float v = sem[m] ? c1[r] : c2[r];
        __builtin_nontemporal_store(v, out + O_X + (size_t)(b0 + m) * DD + n0 + lm);
      }
    }
    // out_e_len = [h | sel](16x512) @ W_L^T(25x512) + b_L : tiles on waves 0,1
    if (wave < 2) {
      int n0 = wave << 4;
      int nrow = n0 + lm; if (nrow > MAXLEN - 1) nrow = MAXLEN - 1;  // clamp OOB reads
      const float* hrow = sm_h + lm * PS + 2 * half;
      const float* srow = (sem[lm] ? sm_ce : sm_pf) + lm * PS + 2 * half;
      const float* bp   = W_L + (size_t)nrow * (HIDN + DD) + 2 * half;
      v8f acc = {};
#pragma unroll 8
      for (int k = 0; k < HIDN + DD; k += 4) {
        const float* ar = (k < HIDN) ? (hrow + k) : (srow + (k - HIDN));
        v2f a = *(const v2f*)ar;
        v2f b = *(const v2f*)(bp + k);
        acc = __builtin_amdgcn_wmma_f32_16x16x4_f32(false, a, false, b,
                                                    (short)0, acc, false, false);
      }
      int col = n0 + lm;
      if (col < MAXLEN) {
        float bb = b_L[col];
#pragma unroll
        for (int r = 0; r < 8; ++r) {
          int m = r + 8 * half;
          out[O_ELEN + (size_t)(b0 + m) * MAXLEN + col] = acc[r] + bb;
        }
      }
    }
  }

  // ---- Phase 5: stream E once (NT) -> scores, d, out_e_idx, E2 copy-out (NT) ----
  for (int p = wave; p < 16 * NE; p += 8) {
    int b = p / NE, e = p - b * NE;
    int bg = b0 + b;
    int e1 = se1[b];
    int em = sem[b];
    int j0 = lane * 8;
    size_t rowoff = ((size_t)bg * NE + e) * DD;

    // prefetch two wave-strides ahead in the E stream
    int pn = p + 16;
    if (pn < 16 * NE) {
      int bn = pn / NE, en = pn - bn * NE;
      __builtin_prefetch(E + ((size_t)(b0 + bn) * NE + en) * DD + j0, 0, 0);
    }

    v4f v0, v1;
    if (e == e1) {
#pragma unroll
      for (int j = 0; j < 4; ++j) {
        v0[j] = sm_ce[b * PS + j0 + j];
        v1[j] = sm_ce[b * PS + j0 + 4 + j];
      }
    } else {
      v0 = nt_load4(E + rowoff + j0);
      v1 = nt_load4(E + rowoff + j0 + 4);
    }
    float dp = 0.f;
#pragma unroll
    for (int j = 0; j < 4; ++j) {
      dp += v0[j] * sm_proj[b * 256 + j0 + j];
      dp += v1[j] * sm_proj[b * 256 + j0 + 4 + j];
    }
    dp = wave_sum(dp);

    nt_store4(out + O_E2 + rowoff + j0,     v0);
    nt_store4(out + O_E2 + rowoff + j0 + 4, v1);

    if (lane == 0) {
      float fin = final_tok[bg] ? 1.f : 0.f;
      float dv = e_dists[(size_t)bg * NE + e] + fin;
      if (e == e1 && em) dv = 0.f;
      if (e == 0) dv = 0.f;
      out[O_D + (size_t)bg * NE + e] = dv;
      bool lk = (e == e1) || (e_lookup[(size_t)bg * (MAXE + 2) + e] != 0);
      float score = dp - __expf(dv * lam0);
      out[O_EIDX + (size_t)bg * NE + e] = lk ? score : -__builtin_inff();
    }
  }
}

extern "C" void kernel_launch(void* const* d_in, const int* in_sizes, int n_in,
                              void* d_out, int out_size, void* d_ws, size_t ws_size,
                              hipStream_t stream) {
  (void)in_sizes; (void)n_in; (void)out_size; (void)d_ws; (void)ws_size;
  const float* h            = (const float*)d_in[0];
  const float* E            = (const float*)d_in[1];
  const int*   n_entities   = (const int*)d_in[2];
  const float* e_dists      = (const float*)d_in[3];
  const float* null_context = (const float*)d_in[4];
  const unsigned char* e_idx_lookup = (const unsigned char*)d_in[5];
  const int*   e_t          = (const int*)d_in[6];
  const int*   e_idx        = (const int*)d_in[7];
  const unsigned char* final_tok = (const unsigned char*)d_in[8];
  const float* init_noise   = (const float*)d_in[9];
  const float* W_R          = (const float*)d_in[10];
  const float* W_ctx        = (const float*)d_in[11];
  const float* lam          = (const float*)d_in[12];
  const float* W_L          = (const float*)d_in[13];
  const float* b_L          = (const float*)d_in[14];
  const float* mean_init    = (const float*)d_in[15];
  const float* W_forget     = (const float*)d_in[16];
  const float* W_input      = (const float*)d_in[17];
  const float* W_X          = (const float*)d_in[18];
  const float* W_Xnull      = (const float*)d_in[19];

  dim3 grid(BB / 16), block(256);
  entity_cell_fused<<<grid, block, 0, stream>>>(
      h, E, n_entities, e_dists, null_context, e_idx_lookup, e_t, e_idx,
      final_tok, init_noise, W_R, W_ctx, lam, W_L, b_L, mean_init,
      W_forget, W_input, W_X, W_Xnull, (float*)d_out);
}